// Encoder_22582938042518
// MI455X (gfx1250) — compile-verified
//
#include <hip/hip_runtime.h>
#include <hip/hip_bf16.h>

typedef __attribute__((ext_vector_type(2))) float v2f;
typedef __attribute__((ext_vector_type(8))) float v8f;

#define IN_C 128
#define NEG_SLOPE 0.2f

// ---------- ordered-float <-> uint monotone mapping for atomic segment-max ----------
__device__ __forceinline__ unsigned f2ord(float f) {
    unsigned u = __float_as_uint(f);
    return (u & 0x80000000u) ? ~u : (u | 0x80000000u);
}
__device__ __forceinline__ float ord2f(unsigned k) {
    unsigned u = (k & 0x80000000u) ? (k & 0x7fffffffu) : ~k;
    return __uint_as_float(u);
}

// ---------------------------------------------------------------------------
// fp32 WMMA GEMM: Y[M,F] = X[M,K] @ W[K,F]
// One wave computes one 16x16 output tile via V_WMMA_F32_16X16X4_F32.
// Block = 128 threads = 4 waves covering 64 rows x 16 cols.
// W column tile (K x 16) staged in LDS once per block.
// ---------------------------------------------------------------------------
__global__ void __launch_bounds__(128) gemm_wmma_f32(
    const float* __restrict__ X, const float* __restrict__ W,
    float* __restrict__ Y, int M, int K, int F)
{
    __shared__ float sW[128 * 16];            // K <= 128
    const int tid = threadIdx.x;
    const int colBase = blockIdx.y * 16;

    for (int i = tid; i < K * 16; i += 128) {
        int k = i >> 4, n = i & 15;
        sW[i] = W[k * F + colBase + n];
    }
    __syncthreads();

    const int wave = tid >> 5;
    const int lane = tid & 31;
    const int rowBase = blockIdx.x * 64 + wave * 16;

    int arow = rowBase + (lane & 15);
    if (arow >= M) arow = M - 1;              // clamp: garbage rows masked at store
    const int khalf = (lane >> 4) << 1;       // lanes 0-15 -> K+0/1, lanes 16-31 -> K+2/3

    v8f c = {};
    for (int k0 = 0; k0 < K; k0 += 4) {
        const float* ap = X + (size_t)arow * K + k0 + khalf;
        v2f a; a.x = ap[0]; a.y = ap[1];
        const float* bp = sW + (k0 + khalf) * 16 + (lane & 15);
        v2f b; b.x = bp[0]; b.y = bp[16];
        c = __builtin_amdgcn_wmma_f32_16x16x4_f32(
                false, a, false, b, (short)0, c, false, false);
    }

    const int rOff = (lane >> 4) << 3;        // lanes 16-31 hold rows +8
    const int col = colBase + (lane & 15);
#pragma unroll
    for (int r = 0; r < 8; ++r) {
        int row = rowBase + rOff + r;
        if (row < M) Y[(size_t)row * F + col] = c[r];
    }
}

// ---------------------------------------------------------------------------
// Attention logits: alpha_s[n,h] = h[n,h,:] . a_src[h,:]   (and alpha_d)
// ---------------------------------------------------------------------------
__global__ void alpha_kernel(const float* __restrict__ h,
                             const float* __restrict__ as_, const float* __restrict__ ad_,
                             float* __restrict__ alS, float* __restrict__ alD,
                             int N, int H, int C)
{
    int idx = blockIdx.x * blockDim.x + threadIdx.x;
    if (idx >= N * H) return;
    int n = idx / H, hd = idx % H;
    const float* hp = h + (size_t)n * H * C + (size_t)hd * C;
    const float* sp = as_ + hd * C;
    const float* dp = ad_ + hd * C;
    float ss = 0.f, dd = 0.f;
    for (int c = 0; c < C; ++c) { float v = hp[c]; ss += v * sp[c]; dd += v * dp[c]; }
    alS[idx] = ss;
    alD[idx] = dd;
}

// ---------------------------------------------------------------------------
// Edge pass 1: e = leaky_relu(alpha_s[src]+alpha_d[dst]); segment max via
// atomicMax on monotone uint key. Edges E..E+N-1 are implicit self loops.
// ---------------------------------------------------------------------------
__global__ void edge_max_kernel(const float* __restrict__ alS, const float* __restrict__ alD,
                                const int* __restrict__ ei, int E, int N, int H,
                                float* __restrict__ ebuf, unsigned* __restrict__ mkey)
{
    int idx = blockIdx.x * blockDim.x + threadIdx.x;
    int tot = (E + N) * H;
    if (idx >= tot) return;
    int e = idx / H, hd = idx % H;
    int s, d;
    if (e < E) { s = ei[e]; d = ei[E + e]; } else { s = d = e - E; }
    float v = alS[s * H + hd] + alD[d * H + hd];
    v = v > 0.f ? v : NEG_SLOPE * v;
    ebuf[idx] = v;
    atomicMax(mkey + d * H + hd, f2ord(v));
}

// Edge pass 2: p = exp(e - m[dst]); segment sum of p
__global__ void edge_exp_kernel(const int* __restrict__ ei, int E, int N, int H,
                                float* __restrict__ ebuf,
                                const unsigned* __restrict__ mkey,
                                float* __restrict__ ssum)
{
    int idx = blockIdx.x * blockDim.x + threadIdx.x;
    int tot = (E + N) * H;
    if (idx >= tot) return;
    int e = idx / H, hd = idx % H;
    int d = (e < E) ? ei[E + e] : (e - E);
    float m = ord2f(mkey[d * H + hd]);
    float p = expf(ebuf[idx] - m);
    ebuf[idx] = p;
    atomicAdd(ssum + d * H + hd, p);
}

// Edge pass 3: one wave32 per (edge, head); lanes stream channels coalesced,
// acc[dst] accumulated with global f32 atomics (working set resides in L2).
__global__ void __launch_bounds__(128) scatter_kernel(
    const float* __restrict__ h, const float* __restrict__ ebuf,
    const float* __restrict__ ssum, const int* __restrict__ ei,
    int E, int N, int H, int C, float* __restrict__ acc)
{
    int wid = blockIdx.x * 4 + (threadIdx.x >> 5);
    int lane = threadIdx.x & 31;
    int tot = (E + N) * H;
    if (wid >= tot) return;
    int e = wid / H, hd = wid % H;
    int s, d;
    if (e < E) { s = ei[e]; d = ei[E + e]; } else { s = d = e - E; }
    float alpha = ebuf[(size_t)e * H + hd] / (ssum[d * H + hd] + 1e-16f);
    const float* hp = h + (size_t)s * H * C + (size_t)hd * C;
    float* ap = acc + (size_t)d * H * C + (size_t)hd * C;
    for (int j = lane; j < C; j += 32)
        atomicAdd(ap + j, hp[j] * alpha);
}

// Mean over heads + bias (+ optional ReLU)
__global__ void finalize_kernel(const float* __restrict__ acc, const float* __restrict__ b,
                                float* __restrict__ xout, int N, int H, int C, int relu)
{
    int idx = blockIdx.x * blockDim.x + threadIdx.x;
    if (idx >= N * C) return;
    int n = idx / C, c = idx % C;
    float s = 0.f;
    for (int hd = 0; hd < H; ++hd) s += acc[(size_t)n * H * C + (size_t)hd * C + c];
    float v = s / (float)H + b[c];
    if (relu) v = fmaxf(v, 0.f);
    xout[idx] = v;
}

// Final heads: z_mean = z@Wm+bm ; z_var = clip(exp(z@Wv+bv), 1e-8, 100)
__global__ void head_kernel(const float* __restrict__ z,
                            const float* __restrict__ Wm, const float* __restrict__ bm,
                            const float* __restrict__ Wv, const float* __restrict__ bv,
                            float* __restrict__ outMean, float* __restrict__ outVar, int N)
{
    int idx = blockIdx.x * blockDim.x + threadIdx.x;
    if (idx >= N * 32) return;
    int n = idx >> 5, j = idx & 31;
    const float* zr = z + (size_t)n * 32;
    float sm = 0.f, sv = 0.f;
    for (int k = 0; k < 32; ++k) {
        float zv = zr[k];
        sm += zv * Wm[k * 32 + j];
        sv += zv * Wv[k * 32 + j];
    }
    sm += bm[j];
    float var = fminf(fmaxf(expf(sv + bv[j]), 1e-8f), 100.f);
    outMean[idx] = sm;
    outVar[idx]  = var;
}

// ---------------------------------------------------------------------------
extern "C" void kernel_launch(void* const* d_in, const int* in_sizes, int n_in,
                              void* d_out, int out_size, void* d_ws, size_t ws_size,
                              hipStream_t stream)
{
    const float* x   = (const float*)d_in[0];
    const int*   ei  = (const int*)d_in[1];   // [2,E] row-major (int32 under default JAX)
    const float* W1  = (const float*)d_in[2];
    const float* as1 = (const float*)d_in[3];
    const float* ad1 = (const float*)d_in[4];
    const float* b1  = (const float*)d_in[5];
    const float* W2  = (const float*)d_in[6];
    const float* as2 = (const float*)d_in[7];
    const float* ad2 = (const float*)d_in[8];
    const float* b2  = (const float*)d_in[9];
    const float* W3  = (const float*)d_in[10];
    const float* as3 = (const float*)d_in[11];
    const float* ad3 = (const float*)d_in[12];
    const float* b3  = (const float*)d_in[13];
    const float* Wm  = (const float*)d_in[14];
    const float* bm  = (const float*)d_in[15];
    const float* Wv  = (const float*)d_in[16];
    const float* bv  = (const float*)d_in[17];

    const int N = in_sizes[0] / IN_C;   // 20000
    const int E = in_sizes[1] / 2;      // 320000
    const int FMAX = 256;               // max H*C

    // ---- workspace carve-up ----
    char* p = (char*)d_ws;
    float*    hA   = (float*)p;    p += (size_t)N * FMAX * sizeof(float);
    float*    acc  = (float*)p;    p += (size_t)N * FMAX * sizeof(float);
    float*    x1   = (float*)p;    p += (size_t)N * 64 * sizeof(float);
    float*    x2   = (float*)p;    p += (size_t)N * 64 * sizeof(float);
    float*    alS  = (float*)p;    p += (size_t)N * 4 * sizeof(float);
    float*    alD  = (float*)p;    p += (size_t)N * 4 * sizeof(float);
    unsigned* mkey = (unsigned*)p; p += (size_t)N * 4 * sizeof(unsigned);
    float*    ssum = (float*)p;    p += (size_t)N * 4 * sizeof(float);
    float*    ebuf = (float*)p;    p += (size_t)(E + N) * 4 * sizeof(float);
    (void)ws_size; (void)n_in; (void)out_size;

    auto run_layer = [&](const float* xin, int K, const float* W,
                         const float* as_, const float* ad_, const float* b,
                         int H, int C, float* xout, int relu)
    {
        const int F = H * C;
        hipMemsetAsync(acc,  0, (size_t)N * F * sizeof(float), stream);
        hipMemsetAsync(mkey, 0, (size_t)N * H * sizeof(unsigned), stream);
        hipMemsetAsync(ssum, 0, (size_t)N * H * sizeof(float), stream);

        dim3 gg((N + 63) / 64, F / 16);
        gemm_wmma_f32<<<gg, 128, 0, stream>>>(xin, W, hA, N, K, F);

        int nh = N * H;
        alpha_kernel<<<(nh + 255) / 256, 256, 0, stream>>>(hA, as_, ad_, alS, alD, N, H, C);

        int te = (E + N) * H;
        edge_max_kernel<<<(te + 255) / 256, 256, 0, stream>>>(alS, alD, ei, E, N, H, ebuf, mkey);
        edge_exp_kernel<<<(te + 255) / 256, 256, 0, stream>>>(ei, E, N, H, ebuf, mkey, ssum);
        scatter_kernel<<<(te + 3) / 4, 128, 0, stream>>>(hA, ebuf, ssum, ei, E, N, H, C, acc);

        int nc = N * C;
        finalize_kernel<<<(nc + 255) / 256, 256, 0, stream>>>(acc, b, xout, N, H, C, relu);
    };

    float* zmean = (float*)d_out;
    float* zvar  = zmean + (size_t)N * 32;
    float* z     = zvar  + (size_t)N * 32;   // third output slice written directly

    run_layer(x,  IN_C, W1, as1, ad1, b1, 4, 64, x1, 1);
    run_layer(x1, 64,   W2, as2, ad2, b2, 4, 64, x2, 1);
    run_layer(x2, 64,   W3, as3, ad3, b3, 1, 32, z,  0);

    head_kernel<<<(N * 32 + 255) / 256, 256, 0, stream>>>(z, Wm, bm, Wv, bv, zmean, zvar, N);
}